// Explicit_corr_55387898249791
// MI455X (gfx1250) — compile-verified
//
#include <hip/hip_runtime.h>
#include <hip/hip_bf16.h>
#include <math.h>

#if defined(__HIP_DEVICE_COMPILE__)
#if !__has_builtin(__builtin_amdgcn_wmma_f32_16x16x4_f32)
#error "missing __builtin_amdgcn_wmma_f32_16x16x4_f32 on device toolchain"
#endif
#endif

typedef __attribute__((ext_vector_type(2))) float v2f;
typedef __attribute__((ext_vector_type(8))) float v8f;

// ---- problem constants ----
#define BATCH 64
#define CHAN  256
#define TS    4          // template spatial
#define CW    20         // candidate spatial
#define NP    289        // 17*17 window offsets
#define FLAT  4624       // 16*289
// output layout (flat f32, concatenated in return order)
static constexpr size_t O_CT = 0;                          // common_template 64*257*16
static constexpr size_t O_CS = 263168;                     // common_search  64*257*4624
static constexpr size_t O_WN = 76318720;                   // weight_norm    64*16
static constexpr size_t O_WC = 76319744;                   // wc_norm        64*4624

// ---------------------------------------------------------------------------
// Per-wave Gram-matrix cosine weight.
// Data: 16 "cells" (M/N = lane%16), 256 channels (K), channel stride in floats.
// A(16xK)=V^T and B(Kx16)=V have identical register images for the f32
// 16x16x4 layout: lane L, vgpr v holds channel (2*(L>>4)+v) of cell (L&15).
// G = V^T V accumulated with V_WMMA_F32_16X16X4_F32 (64 instructions).
// Then: dot[k]=colsum/16, ||v_k||=sqrt(diag), ||mean||=sqrt(sum(G))/16.
// Returns the cosine weight for cell n=lane&15 (duplicated on both halves).
// ---------------------------------------------------------------------------
__device__ __forceinline__ float gram_weight(const float* __restrict__ base, int chanStride) {
    const int lane = threadIdx.x & 31;
    const int half = lane >> 4;
    const int n    = lane & 15;

    v8f acc = {0.f, 0.f, 0.f, 0.f, 0.f, 0.f, 0.f, 0.f};
    const float* p = base + (size_t)(2 * half) * chanStride;
    #pragma unroll 4
    for (int c0 = 0; c0 < CHAN; c0 += 4) {
        v2f ab;
        ab.x = p[0];
        ab.y = p[chanStride];
        p += (size_t)4 * chanStride;
        acc = __builtin_amdgcn_wmma_f32_16x16x4_f32(
            /*neg_a=*/false, ab, /*neg_b=*/false, ab,
            /*c_mod=*/(short)0, acc, /*reuse_a=*/false, /*reuse_b=*/false);
    }

    // this lane holds G[M = d + 8*half, N = n] for d=0..7
    float partial = acc[0] + acc[1] + acc[2] + acc[3] +
                    acc[4] + acc[5] + acc[6] + acc[7];
    float colsum = partial + __shfl_xor(partial, 16, 32);     // sum_M G[M,n]
    float qp = ((n >> 3) == half) ? acc[n & 7] : 0.0f;        // diag G[n,n]
    float Q  = qp + __shfl_xor(qp, 16, 32);
    float s2 = partial;                                       // sum of all G
    #pragma unroll
    for (int m = 1; m < 32; m <<= 1) s2 += __shfl_xor(s2, m, 32);

    float dotv = colsum * 0.0625f;          // mean = s/16
    float na   = sqrtf(s2) * 0.0625f;       // ||mean||
    float nb   = sqrtf(Q);                  // ||v_n||
    return dotv / (fmaxf(na, 1e-8f) * fmaxf(nb, 1e-8f));
}

// ---- kernel 1: raw candidate weights (staged into the wc_norm region) ----
__global__ __launch_bounds__(32) void k_weight_cand(const float* __restrict__ cand,
                                                    float* out) {
    unsigned bid = blockIdx.x;
    int b  = bid / NP;
    int pp = bid % NP;
    int pi = pp / 17, pj = pp % 17;
    int n  = threadIdx.x & 15;
    int y  = pi + (n >> 2), x = pj + (n & 3);
    const float* base = cand + (size_t)b * CHAN * (CW * CW) + (size_t)(y * CW + x);
    float w = gram_weight(base, CW * CW);
    if ((threadIdx.x >> 4) == 0)
        out[O_WC + (size_t)b * FLAT + (size_t)n * NP + pp] = w;   // flat [ki,kj,pi,pj]
}

// ---- kernel 2: per-batch softmax(4624) + permuted scatter -----------------
// soft flat f = n*289+p. Reference writes wc_norm[r*289+q] = soft[q*16+r]
// and common_search[:,256,r,q] = soft[q*16+r]. Stage in LDS, overwrite in place.
__global__ __launch_bounds__(256) void k_softmax_cand(float* out) {
    int b = blockIdx.x, tid = threadIdx.x;
    float* wreg = out + O_WC + (size_t)b * FLAT;
    __shared__ float sw[FLAT];
    __shared__ float red[8];

    for (int g = tid; g < FLAT; g += 256) sw[g] = wreg[g];
    __syncthreads();

    float mx = -3.4e38f;
    for (int g = tid; g < FLAT; g += 256) mx = fmaxf(mx, sw[g]);
    #pragma unroll
    for (int m = 1; m < 32; m <<= 1) mx = fmaxf(mx, __shfl_xor(mx, m, 32));
    if ((tid & 31) == 0) red[tid >> 5] = mx;
    __syncthreads();
    mx = red[0];
    #pragma unroll
    for (int i = 1; i < 8; i++) mx = fmaxf(mx, red[i]);
    __syncthreads();

    float s = 0.f;
    for (int g = tid; g < FLAT; g += 256) s += expf(sw[g] - mx);
    #pragma unroll
    for (int m = 1; m < 32; m <<= 1) s += __shfl_xor(s, m, 32);
    if ((tid & 31) == 0) red[tid >> 5] = s;
    __syncthreads();
    s = 0.f;
    #pragma unroll
    for (int i = 0; i < 8; i++) s += red[i];
    float invs = 1.0f / s;

    for (int g = tid; g < FLAT; g += 256) {
        float v = expf(sw[g] - mx) * invs;
        int q = g >> 4, r = g & 15;                 // g = q*16 + r
        out[O_WC + (size_t)b * FLAT + (size_t)r * NP + q] = v;
        out[O_CS + (((size_t)b * 257 + 256) * 16 + r) * NP + q] = v;
    }
}

// ---- kernel 3: template weights + softmax(16) + channel-256 write ---------
__global__ __launch_bounds__(32) void k_template(const float* __restrict__ tmpl,
                                                 float* out) {
    int b = blockIdx.x;
    int n = threadIdx.x & 15;
    const float* base = tmpl + (size_t)b * CHAN * (TS * TS)
                             + (size_t)((n >> 2) * TS + (n & 3));
    float w = gram_weight(base, TS * TS);
    float mx = w;
    #pragma unroll
    for (int m = 1; m < 32; m <<= 1) mx = fmaxf(mx, __shfl_xor(mx, m, 32));
    float e = expf(w - mx);
    float s = e;
    #pragma unroll
    for (int m = 1; m < 32; m <<= 1) s += __shfl_xor(s, m, 32);
    s *= 0.5f;                                   // both halves hold duplicates
    float wn = e / s;
    if ((threadIdx.x >> 4) == 0) {
        out[O_WN + (size_t)b * 16 + n] = wn;
        out[O_CT + ((size_t)b * 257 + 256) * 16 + n] = wn;
    }
}

// ---- kernel 4: copy template into common_template channels 0..255 ---------
__global__ __launch_bounds__(256) void k_copy_template(const float* __restrict__ tmpl,
                                                       float* __restrict__ out) {
    unsigned i = blockIdx.x * 256u + threadIdx.x;   // N = 64*256*16 = 262144 exact
    unsigned k = i & 15u, t = i >> 4;
    unsigned c = t & 255u, b = t >> 8;
    out[O_CT + (((size_t)b * 257 + c) << 4) + k] = tmpl[i];
}

// ---- kernel 5: big sliding-window gather into common_search ---------------
__global__ __launch_bounds__(256) void k_gather(const float* __restrict__ cand,
                                                float* __restrict__ out) {
    const unsigned N = 75759616u;                   // 64*256*16*289
    for (unsigned i = blockIdx.x * 256u + threadIdx.x; i < N;
         i += gridDim.x * 256u) {
        unsigned q = i % NP;  unsigned t = i / NP;
        unsigned k = t & 15u; unsigned t2 = t >> 4;
        unsigned c = t2 & 255u; unsigned b = t2 >> 8;
        unsigned pi = q / 17u, pj = q % 17u;
        unsigned y = (k >> 2) + pi, x = (k & 3) + pj;
        out[O_CS + ((((size_t)b * 257 + c) << 4) + k) * NP + q] =
            cand[(((size_t)b << 8) + c) * (CW * CW) + y * CW + x];
    }
}

extern "C" void kernel_launch(void* const* d_in, const int* in_sizes, int n_in,
                              void* d_out, int out_size, void* d_ws, size_t ws_size,
                              hipStream_t stream) {
    const float* tmpl = (const float*)d_in[0];   // (64,256,4,4)
    const float* cand = (const float*)d_in[1];   // (64,256,20,20)
    float* out = (float*)d_out;
    (void)in_sizes; (void)n_in; (void)out_size; (void)d_ws; (void)ws_size;

    k_weight_cand <<<BATCH * NP, 32, 0, stream>>>(cand, out);
    k_softmax_cand<<<BATCH,     256, 0, stream>>>(out);
    k_template    <<<BATCH,      32, 0, stream>>>(tmpl, out);
    k_copy_template<<<1024,     256, 0, stream>>>(tmpl, out);
    k_gather      <<<4096,      256, 0, stream>>>(cand, out);
}